// TGCN_11742440587919
// MI455X (gfx1250) — compile-verified
//
#include <hip/hip_runtime.h>

typedef __attribute__((ext_vector_type(2))) float v2f;
typedef __attribute__((ext_vector_type(8))) float v8f;

// ---------------------------------------------------------------- degree prep
__global__ void k_fill1(float* __restrict__ p, int n) {
    int i = blockIdx.x * blockDim.x + threadIdx.x;
    if (i < n) p[i] = 1.0f;  // self-loop contributes 1 to every degree
}

__global__ void k_deg_scatter(float* __restrict__ deg, const int* __restrict__ dst, int nE) {
    int e = blockIdx.x * blockDim.x + threadIdx.x;
    if (e < nE) atomicAdd(&deg[dst[e]], 1.0f);
}

__global__ void k_rsqrt_inplace(float* __restrict__ p, int n) {
    int i = blockIdx.x * blockDim.x + threadIdx.x;
    if (i < n) p[i] = rsqrtf(p[i]);  // deg >= 1 always (self-loops)
}

// ------------------------------------------------- small / degenerate GEMMs
// Y[i][j] = dinv[i] * sum_k X[i][k] * W[k][j];  AGG initialized = Y (self loop)
__global__ void k_gemm_small(const float* __restrict__ X, const float* __restrict__ W,
                             const float* __restrict__ dinv,
                             float* __restrict__ Y, float* __restrict__ AGG,
                             int n, int kin, int kout) {
    int idx = blockIdx.x * blockDim.x + threadIdx.x;
    if (idx >= n * kout) return;
    int i = idx / kout;
    int j = idx - i * kout;
    const float* xr = X + (size_t)i * kin;
    const float* wc = W + j;
    float acc = 0.0f;
    for (int k = 0; k < kin; ++k) acc = fmaf(xr[k], wc[(size_t)k * kout], acc);
    float v = acc * dinv[i];
    Y[idx] = v;
    AGG[idx] = v;
}

// --------------------------------------------- layer-2 GEMM: f32 WMMA 16x16x4
// A: [M,256] row-major (H1), B: [256,128] row-major (W2). One wave per 16x16
// output tile; 64 k-steps of V_WMMA_F32_16X16X4_F32 (exact f32, matches ref).
__global__ void __launch_bounds__(256)
k_gemm_wmma_k256_n128(const float* __restrict__ A, const float* __restrict__ B,
                      const float* __restrict__ dinv,
                      float* __restrict__ Y, float* __restrict__ AGG) {
    const int wave = threadIdx.x >> 5;   // 8 waves -> 8 N-tiles (N=128)
    const int lane = threadIdx.x & 31;
    const int hi   = lane >> 4;          // selects K pair (A) / K row pair (B)
    const int mn   = lane & 15;          // M row (A) / N col (B,C,D)
    const int mbase = blockIdx.x * 16;   // 1250 blocks cover M=20000 exactly
    const int nbase = wave * 16;

    const float* Arow = A + (size_t)(mbase + mn) * 256;
    v8f acc = {};
#pragma unroll 8
    for (int k = 0; k < 256; k += 4) {
        const int ka = k + 2 * hi;
        v2f a, b;
        a.x = Arow[ka];                  // per-lane contiguous 64-bit load
        a.y = Arow[ka + 1];
        const float* Bp = B + (size_t)ka * 128 + nbase + mn;
        b.x = Bp[0];                     // coalesced across lanes
        b.y = Bp[128];
        acc = __builtin_amdgcn_wmma_f32_16x16x4_f32(
            /*neg_a=*/false, a, /*neg_b=*/false, b,
            /*c_mod=*/(short)0, acc, /*reuse_a=*/false, /*reuse_b=*/false);
    }
#pragma unroll
    for (int r = 0; r < 8; ++r) {        // D layout: row = r + 8*hi, col = mn
        const int row = mbase + r + 8 * hi;
        const int col = nbase + mn;
        float v = acc[r] * dinv[row];
        Y[(size_t)row * 128 + col] = v;
        AGG[(size_t)row * 128 + col] = v;
    }
}

// ----------------------------------------------------------- edge aggregation
// One thread per (edge, 4-feature chunk). Working set fits in 192MB L2, so
// gathers + f32 atomics resolve in L2 -> L2-bandwidth bound, which is optimal
// for an unsorted edge list.
__global__ void k_scatter_vec4(const float* __restrict__ Y, float* __restrict__ AGG,
                               const int* __restrict__ src, const int* __restrict__ dst,
                               int nE, int c4shift /* feat/4 = 1<<c4shift */) {
    int idx = blockIdx.x * blockDim.x + threadIdx.x;
    if (idx >= (nE << c4shift)) return;
    int e = idx >> c4shift;
    int c = (idx & ((1 << c4shift) - 1)) << 2;
    int feat = 4 << c4shift;
    int s = src[e], d = dst[e];
    const float4 v = *(const float4*)(Y + (size_t)s * feat + c);
    float* p = AGG + (size_t)d * feat + c;
    atomicAdd(p + 0, v.x);
    atomicAdd(p + 1, v.y);
    atomicAdd(p + 2, v.z);
    atomicAdd(p + 3, v.w);
}

__global__ void k_scatter_scalar(const float* __restrict__ Y, float* __restrict__ AGG,
                                 const int* __restrict__ src, const int* __restrict__ dst,
                                 int nE) {
    int e = blockIdx.x * blockDim.x + threadIdx.x;
    if (e < nE) atomicAdd(&AGG[dst[e]], Y[src[e]]);
}

// --------------------------------------------------------------- layer finish
// H = act(dinv[i]*AGG + b[j])
__global__ void k_finish(const float* __restrict__ AGG, const float* __restrict__ dinv,
                         const float* __restrict__ b, float* __restrict__ H,
                         int n, int kout, int do_relu) {
    int idx = blockIdx.x * blockDim.x + threadIdx.x;
    if (idx >= n * kout) return;
    int i = idx / kout;
    int j = idx - i * kout;
    float v = fmaf(dinv[i], AGG[idx], b[j]);
    if (do_relu) v = fmaxf(v, 0.0f);
    H[idx] = v;
}

// ------------------------------------------------------------------- launcher
extern "C" void kernel_launch(void* const* d_in, const int* in_sizes, int n_in,
                              void* d_out, int out_size, void* d_ws, size_t ws_size,
                              hipStream_t stream) {
    const float* x  = (const float*)d_in[0];   // [N,10]
    const int*   ei = (const int*)d_in[1];     // [2,E]
    const float* W1 = (const float*)d_in[2];   // [10,256]
    const float* b1 = (const float*)d_in[3];
    const float* W2 = (const float*)d_in[4];   // [256,128]
    const float* b2 = (const float*)d_in[5];
    const float* W3 = (const float*)d_in[6];   // [128,1]
    const float* b3 = (const float*)d_in[7];

    const int N = in_sizes[0] / 10;
    const int E = in_sizes[1] / 2;
    const int* src = ei;
    const int* dst = ei + E;

    // workspace layout (floats); H1 aliases Y1, H2 aliases Y2 (dead after scatter)
    float* ws   = (float*)d_ws;
    float* dinv = ws;                          size_t off = (size_t)N;
    float* Y1   = ws + off;                    off += (size_t)N * 256;
    float* AGG1 = ws + off;                    off += (size_t)N * 256;
    float* Y2   = ws + off;                    off += (size_t)N * 128;
    float* AGG2 = ws + off;                    off += (size_t)N * 128;
    float* Y3   = ws + off;                    off += (size_t)N;
    float* AGG3 = ws + off;                    off += (size_t)N;
    float* H1 = Y1;
    float* H2 = Y2;
    (void)ws_size; (void)n_in; (void)out_size;

    const int T = 256;
    // --- degrees -> dinv (reused by all 3 layers)
    k_fill1<<<(N + T - 1) / T, T, 0, stream>>>(dinv, N);
    k_deg_scatter<<<(E + T - 1) / T, T, 0, stream>>>(dinv, dst, E);
    k_rsqrt_inplace<<<(N + T - 1) / T, T, 0, stream>>>(dinv, N);

    // --- layer 1: [N,10]@[10,256], relu
    k_gemm_small<<<((size_t)N * 256 + T - 1) / T, T, 0, stream>>>(x, W1, dinv, Y1, AGG1, N, 10, 256);
    k_scatter_vec4<<<(((size_t)E << 6) + T - 1) / T, T, 0, stream>>>(Y1, AGG1, src, dst, E, 6);
    k_finish<<<((size_t)N * 256 + T - 1) / T, T, 0, stream>>>(AGG1, dinv, b1, H1, N, 256, 1);

    // --- layer 2: [N,256]@[256,128] via V_WMMA_F32_16X16X4_F32, relu
    k_gemm_wmma_k256_n128<<<N / 16, 256, 0, stream>>>(H1, W2, dinv, Y2, AGG2);
    k_scatter_vec4<<<(((size_t)E << 5) + T - 1) / T, T, 0, stream>>>(Y2, AGG2, src, dst, E, 5);
    k_finish<<<((size_t)N * 128 + T - 1) / T, T, 0, stream>>>(AGG2, dinv, b2, H2, N, 128, 1);

    // --- layer 3: [N,128]@[128,1], no relu, write d_out
    k_gemm_small<<<(N + T - 1) / T, T, 0, stream>>>(H2, W3, dinv, Y3, AGG3, N, 128, 1);
    k_scatter_scalar<<<(E + T - 1) / T, T, 0, stream>>>(Y3, AGG3, src, dst, E);
    k_finish<<<(N + T - 1) / T, T, 0, stream>>>(AGG3, dinv, b3, (float*)d_out, N, 1, 0);
}